// InternalSequenceEmbedder_41034117546191
// MI455X (gfx1250) — compile-verified
//
#include <hip/hip_runtime.h>
#include <math.h>

// Problem constants (match reference)
#define WB 32
#define WT 8192
#define WE 256
#define WD 32
#define WH 32
#define WM 64
#define CHUNK 512
#define NCH (WT / CHUNK)          // 16 chunks
#define NROWS (WB * WT)           // 262144 (b,t) rows
#define NTILES (NROWS / 16)       // 16384 16-row WMMA tiles
#define WAVES 8                   // waves per workgroup (256 threads, wave32)

typedef __attribute__((ext_vector_type(2))) float v2f;
typedef __attribute__((ext_vector_type(4))) float v4f;
typedef __attribute__((ext_vector_type(8))) float v8f;

// D(16x16,f32) = A(16x4,f32) x B(4x16,f32) + C  -- CDNA5 fp32 WMMA
__device__ __forceinline__ v8f wmma4(v2f a, v2f b, v8f c) {
  return __builtin_amdgcn_wmma_f32_16x16x4_f32(
      /*neg_a=*/false, a, /*neg_b=*/false, b,
      /*c_mod=*/(short)0, c, /*reuse_a=*/false, /*reuse_b=*/false);
}

// A-frag (16x4 f32): lane L holds row M=L%16; K = kk + 2*(L/16) + {0,1}
__device__ __forceinline__ v2f afrag(const float* s, int pitch, int kk, int lane) {
  const float* q = s + (lane & 15) * pitch + kk + ((lane >> 4) << 1);
  v2f a; a[0] = q[0]; a[1] = q[1]; return a;
}
// B-frag (4x16 f32) from K-pair-interleaved weights: element [k/2][col] is the
// float2 {W[k][col], W[k+1][col]} -> one ds_load_b64 per fragment, no v_mov packing.
// lane L: col N=(L%16)+noff ; pair row = kk/2 + (L/16)   (kk is a multiple of 4)
__device__ __forceinline__ v2f bfrag2(const float* w, int pairpitch, int kk, int noff, int lane) {
  int kp = (kk >> 1) + (lane >> 4);
  int c  = (lane & 15) + noff;
  return *((const v2f*)w + kp * pairpitch + c);
}
// C-tile (16x16 f32) -> LDS row-major [M][noff+N]: reg r is row r + 8*(L/16)
__device__ __forceinline__ void cstore(float* s, int pitch, int noff, int lane, v8f c) {
  int cc = (lane & 15) + noff;
  int rb = (lane >> 4) << 3;
#pragma unroll
  for (int r = 0; r < 8; ++r) s[(rb + r) * pitch + cc] = c[r];
}

__device__ __forceinline__ float sigmoidf_(float x) { return 1.f / (1.f + expf(-x)); }
__device__ __forceinline__ float gelu_tanh(float x) {  // jax.nn.gelu (approximate=True)
  float x3 = x * x * x;
  return 0.5f * x * (1.f + tanhf(0.7978845608028654f * (x + 0.044715f * x3)));
}

// Issue one 16x64 f32 panel as 8 per-lane async global->LDS b128 copies (ASYNCcnt).
__device__ __forceinline__ void issue_panel(const float* __restrict__ e, long n0, int p,
                                            float* Pbuf, int lane) {
  unsigned lbase =
      (unsigned)(unsigned long long)(__attribute__((address_space(3))) float*)Pbuf;
#pragma unroll
  for (int i = 0; i < 8; ++i) {
    int j = lane + 32 * i;                 // 0..255 float4s of the panel
    int row = j >> 4, c4 = j & 15;
    const float* g = e + (n0 + row) * WE + p * 64 + c4 * 4;
    unsigned loff = lbase + (unsigned)((row * 68 + c4 * 4) * 4);
    asm volatile("global_load_async_to_lds_b128 %0, %1, off"
                 :: "v"(loff), "v"(g) : "memory");
  }
}

// ---------------------------------------------------------------------------
// K1: x = e@Wpi + bpi ; u_f = x@fWin + fbin ; u_b = x@bWin + bbin
// e streamed via double-buffered async global->LDS panels, per wave.
// ---------------------------------------------------------------------------
__global__ __launch_bounds__(256) void k1_proj_u(
    const float* __restrict__ e, const float* __restrict__ Wpi, const float* __restrict__ bpi,
    const float* __restrict__ fWin, const float* __restrict__ fbin,
    const float* __restrict__ bWin, const float* __restrict__ bbin,
    float* __restrict__ uf, float* __restrict__ ub)
{
  __shared__ __align__(16) float sWpi[128 * 33 * 2];   // K-pair interleaved, pair pitch 33
  __shared__ __align__(16) float sWf[16 * 33 * 2];
  __shared__ __align__(16) float sWb[16 * 33 * 2];
  __shared__ float sb[3][WD];
  __shared__ __align__(16) float pan[WAVES][2][16 * 68]; // per-wave double-buffered panels
  __shared__ __align__(16) float xs_[WAVES][16 * 36];    // per-wave 16x32 scratch, pitch 36

  int tid = threadIdx.x;
  for (int i = tid; i < WE * WD; i += 256) {
    int k = i >> 5, c = i & 31;
    sWpi[((k >> 1) * 33 + c) * 2 + (k & 1)] = Wpi[i];
  }
  for (int i = tid; i < WD * WD; i += 256) {
    int k = i >> 5, c = i & 31;
    int r = ((k >> 1) * 33 + c) * 2 + (k & 1);
    sWf[r] = fWin[i]; sWb[r] = bWin[i];
  }
  if (tid < WD) { sb[0][tid] = bpi[tid]; sb[1][tid] = fbin[tid]; sb[2][tid] = bbin[tid]; }
  __syncthreads();

  int wave = tid >> 5, lane = tid & 31;
  long n0 = ((long)blockIdx.x * WAVES + wave) * 16;     // first (b,t) row of tile
  int col = lane & 15;
  float* Pb0 = pan[wave][0];
  float* Pb1 = pan[wave][1];
  float* X   = xs_[wave];

  v8f x0, x1;
  { float b0 = sb[0][col], b1 = sb[0][col + 16];
#pragma unroll
    for (int i = 0; i < 8; ++i) { x0[i] = b0; x1[i] = b1; } }

  issue_panel(e, n0, 0, Pb0, lane);                     // prefetch panel 0
#pragma unroll
  for (int p = 0; p < 4; ++p) {                         // K panels of 64 over E=256
    float* P = (p & 1) ? Pb1 : Pb0;
    if (p < 3) {
      issue_panel(e, n0, p + 1, (p & 1) ? Pb0 : Pb1, lane);  // prefetch next panel
      asm volatile("s_wait_asynccnt 0x8" ::: "memory");      // panel p landed (in-order)
    } else {
      asm volatile("s_wait_asynccnt 0x0" ::: "memory");
    }
    __builtin_amdgcn_wave_barrier();
#pragma unroll
    for (int kk = 0; kk < 64; kk += 4) {
      v2f a = afrag(P, 68, kk, lane);
      int kg = p * 64 + kk;
      x0 = wmma4(a, bfrag2(sWpi, 33, kg, 0, lane), x0);
      x1 = wmma4(a, bfrag2(sWpi, 33, kg, 16, lane), x1);
    }
    __builtin_amdgcn_wave_barrier();
    asm volatile("s_wait_dscnt 0x0" ::: "memory");      // panel reads done before reuse
  }

  cstore(X, 36, 0, lane, x0);                           // x tile -> LDS (C->A relayout)
  cstore(X, 36, 16, lane, x1);
  __builtin_amdgcn_wave_barrier();

  v8f u_f0, u_f1, u_b0, u_b1;
  { float a0 = sb[1][col], a1 = sb[1][col + 16], c0 = sb[2][col], c1 = sb[2][col + 16];
#pragma unroll
    for (int i = 0; i < 8; ++i) { u_f0[i] = a0; u_f1[i] = a1; u_b0[i] = c0; u_b1[i] = c1; } }
#pragma unroll
  for (int kk = 0; kk < 32; kk += 4) {
    v2f a = afrag(X, 36, kk, lane);
    u_f0 = wmma4(a, bfrag2(sWf, 33, kk, 0, lane), u_f0);
    u_f1 = wmma4(a, bfrag2(sWf, 33, kk, 16, lane), u_f1);
    u_b0 = wmma4(a, bfrag2(sWb, 33, kk, 0, lane), u_b0);
    u_b1 = wmma4(a, bfrag2(sWb, 33, kk, 16, lane), u_b1);
  }
  __builtin_amdgcn_wave_barrier();

  cstore(X, 36, 0, lane, u_f0);                         // stage + coalesced b128 stores
  cstore(X, 36, 16, lane, u_f1);
  __builtin_amdgcn_wave_barrier();
#pragma unroll
  for (int i = 0; i < 4; ++i) {
    int j = lane + 32 * i; int row = j >> 3, c4 = j & 7;
    *((v4f*)(uf + n0 * WH) + j) = *((const v4f*)(X + row * 36) + c4);
  }
  __builtin_amdgcn_wave_barrier();
  cstore(X, 36, 0, lane, u_b0);
  cstore(X, 36, 16, lane, u_b1);
  __builtin_amdgcn_wave_barrier();
#pragma unroll
  for (int i = 0; i < 4; ++i) {
    int j = lane + 32 * i; int row = j >> 3, c4 = j & 7;
    *((v4f*)(ub + n0 * WH) + j) = *((const v4f*)(X + row * 36) + c4);
  }
}

// ---------------------------------------------------------------------------
// K2a: in-place local scans on 512-long chunks (fwd over uf, bwd over ub).
// ---------------------------------------------------------------------------
__global__ void k2a_localscan(const float* __restrict__ floga, const float* __restrict__ bloga,
                              float* __restrict__ uf, float* __restrict__ ub,
                              float* __restrict__ carry)
{
  int tid = blockIdx.x * blockDim.x + threadIdx.x;      // 0..32767
  int hh = tid & 31;
  int c  = (tid >> 5) & (NCH - 1);
  int b  = (tid >> 9) & 31;
  int d  = tid >> 14;
  float a = sigmoidf_(d ? bloga[hh] : floga[hh]);
  float* u = d ? ub : uf;
  float s = 0.f;
  long base = d ? (((long)b * WT + (long)c * CHUNK + CHUNK - 1) * WH + hh)
                : (((long)b * WT + (long)c * CHUNK) * WH + hh);
  long step = d ? -(long)WH : (long)WH;
#pragma unroll 4
  for (int j = 0; j < CHUNK; ++j) { s = a * s + u[base]; u[base] = s; base += step; }
  carry[((d * 32 + b) * NCH + c) * 32 + hh] = s;
}

// K2b: exclusive scan of chunk carries -> state entering each chunk (in place).
__global__ void k2b_chunkprefix(const float* __restrict__ floga, const float* __restrict__ bloga,
                                float* __restrict__ carry)
{
  int tid = blockIdx.x * blockDim.x + threadIdx.x;      // 0..2047
  int hh = tid & 31;
  int b  = (tid >> 5) & 31;
  int d  = tid >> 10;
  float a  = sigmoidf_(d ? bloga[hh] : floga[hh]);
  float aL = powf(a, (float)CHUNK);
  float pref = 0.f;
  if (d == 0) {
    for (int c = 0; c < NCH; ++c) {
      int idx = ((d * 32 + b) * NCH + c) * 32 + hh;
      float cv = carry[idx]; carry[idx] = pref; pref = aL * pref + cv;
    }
  } else {
    for (int c = NCH - 1; c >= 0; --c) {
      int idx = ((d * 32 + b) * NCH + c) * 32 + hh;
      float cv = carry[idx]; carry[idx] = pref; pref = aL * pref + cv;
    }
  }
}

// K2c: h[t] += a^(j+1) * prefix  (block = one (dir,b,chunk), lane = h)
__global__ void k2c_apply(const float* __restrict__ floga, const float* __restrict__ bloga,
                          float* __restrict__ uf, float* __restrict__ ub,
                          const float* __restrict__ carry)
{
  int bx = blockIdx.x;                                  // 0..1023
  int hh = threadIdx.x;                                 // 0..31
  int c  = bx & (NCH - 1);
  int b  = (bx >> 4) & 31;
  int d  = bx >> 9;
  if ((d == 0 && c == 0) || (d == 1 && c == NCH - 1)) return;  // zero prefix
  float a = sigmoidf_(d ? bloga[hh] : floga[hh]);
  float pref = carry[((d * 32 + b) * NCH + c) * 32 + hh];
  float* u = d ? ub : uf;
  long base = d ? (((long)b * WT + (long)c * CHUNK + CHUNK - 1) * WH + hh)
                : (((long)b * WT + (long)c * CHUNK) * WH + hh);
  long step = d ? -(long)WH : (long)WH;
  float pw = a;
#pragma unroll 4
  for (int j = 0; j < CHUNK; ++j) { u[base] += pw * pref; pw *= a; base += step; }
}

// ---------------------------------------------------------------------------
// K3: per direction: y = h@Wout+bout ; y + gelu(y@W1+b1)@W2 + b2 ; then
//     out = [fwd,bwd]@Wpo + bpo.  One 16-row tile per wave, all WMMA.
// ---------------------------------------------------------------------------
__global__ __launch_bounds__(256) void k3_head(
    const float* __restrict__ hf, const float* __restrict__ hb,
    const float* __restrict__ fWout, const float* __restrict__ fbout,
    const float* __restrict__ fW1, const float* __restrict__ fb1,
    const float* __restrict__ fW2, const float* __restrict__ fb2,
    const float* __restrict__ bWout, const float* __restrict__ bbout,
    const float* __restrict__ bW1, const float* __restrict__ bb1,
    const float* __restrict__ bW2, const float* __restrict__ bb2,
    const float* __restrict__ Wpo, const float* __restrict__ bpo,
    float* __restrict__ out)
{
  __shared__ __align__(16) float sWo[2][16 * 33 * 2];   // K-pair interleaved layouts
  __shared__ __align__(16) float sW1[2][16 * 65 * 2];
  __shared__ __align__(16) float sW2[2][32 * 33 * 2];
  __shared__ __align__(16) float sWp[32 * 33 * 2];
  __shared__ float sbo[2][WD], sb1v[2][WM], sb2v[2][WD], sbp[WD];
  __shared__ __align__(16) float S[WAVES][16 * 68];     // per-wave 16x64 scratch, pitch 68

  int tid = threadIdx.x;
  for (int i = tid; i < WD * WD; i += 256) {
    int k = i >> 5, c = i & 31;
    int r = ((k >> 1) * 33 + c) * 2 + (k & 1);
    sWo[0][r] = fWout[i]; sWo[1][r] = bWout[i];
  }
  for (int i = tid; i < WD * WM; i += 256) {
    int k = i >> 6, c = i & 63;
    int r = ((k >> 1) * 65 + c) * 2 + (k & 1);
    sW1[0][r] = fW1[i]; sW1[1][r] = bW1[i];
  }
  for (int i = tid; i < WM * WD; i += 256) {
    int k = i >> 5, c = i & 31;
    int r = ((k >> 1) * 33 + c) * 2 + (k & 1);
    sW2[0][r] = fW2[i]; sW2[1][r] = bW2[i]; sWp[r] = Wpo[i];
  }
  if (tid < WM) { sb1v[0][tid] = fb1[tid]; sb1v[1][tid] = bb1[tid]; }
  if (tid < WD) {
    sbo[0][tid] = fbout[tid]; sbo[1][tid] = bbout[tid];
    sb2v[0][tid] = fb2[tid];  sb2v[1][tid] = bb2[tid];
    sbp[tid] = bpo[tid];
  }
  __syncthreads();

  int wave = tid >> 5, lane = tid & 31;
  long n0 = ((long)blockIdx.x * WAVES + wave) * 16;
  int col = lane & 15;
  float* Sw = S[wave];

  v8f f0, f1, m0, m1;
  for (int d = 0; d < 2; ++d) {
    const float* h = d ? hb : hf;
#pragma unroll
    for (int i = 0; i < 4; ++i) {                       // coalesced 16x32 tile load
      int j = lane + 32 * i; int row = j >> 3, c4 = j & 7;
      *((v4f*)(Sw + row * 68) + c4) = *((const v4f*)(h + n0 * WH) + j);
    }
    __builtin_amdgcn_wave_barrier();

    v8f y0, y1;
    { float b0 = sbo[d][col], b1 = sbo[d][col + 16];
#pragma unroll
      for (int i = 0; i < 8; ++i) { y0[i] = b0; y1[i] = b1; } }
#pragma unroll
    for (int kk = 0; kk < 32; kk += 4) {                // y = h@Wout
      v2f a = afrag(Sw, 68, kk, lane);
      y0 = wmma4(a, bfrag2(sWo[d], 33, kk, 0, lane), y0);
      y1 = wmma4(a, bfrag2(sWo[d], 33, kk, 16, lane), y1);
    }
    __builtin_amdgcn_wave_barrier();
    cstore(Sw, 68, 0, lane, y0);
    cstore(Sw, 68, 16, lane, y1);
    __builtin_amdgcn_wave_barrier();

    v8f t0, t1, t2, t3;
    { float b0 = sb1v[d][col], b1 = sb1v[d][col + 16],
            b2 = sb1v[d][col + 32], b3 = sb1v[d][col + 48];
#pragma unroll
      for (int i = 0; i < 8; ++i) { t0[i] = b0; t1[i] = b1; t2[i] = b2; t3[i] = b3; } }
#pragma unroll
    for (int kk = 0; kk < 32; kk += 4) {                // t = y@W1 (N=64)
      v2f a = afrag(Sw, 68, kk, lane);
      t0 = wmma4(a, bfrag2(sW1[d], 65, kk, 0,  lane), t0);
      t1 = wmma4(a, bfrag2(sW1[d], 65, kk, 16, lane), t1);
      t2 = wmma4(a, bfrag2(sW1[d], 65, kk, 32, lane), t2);
      t3 = wmma4(a, bfrag2(sW1[d], 65, kk, 48, lane), t3);
    }
#pragma unroll
    for (int i = 0; i < 8; ++i) {                       // elementwise tanh-GELU
      t0[i] = gelu_tanh(t0[i]); t1[i] = gelu_tanh(t1[i]);
      t2[i] = gelu_tanh(t2[i]); t3[i] = gelu_tanh(t3[i]);
    }
    __builtin_amdgcn_wave_barrier();
    cstore(Sw, 68, 0, lane, t0);  cstore(Sw, 68, 16, lane, t1);
    cstore(Sw, 68, 32, lane, t2); cstore(Sw, 68, 48, lane, t3);
    __builtin_amdgcn_wave_barrier();

    v8f g0, g1;
    { float b0 = sb2v[d][col], b1 = sb2v[d][col + 16];
#pragma unroll
      for (int i = 0; i < 8; ++i) { g0[i] = b0; g1[i] = b1; } }
#pragma unroll
    for (int kk = 0; kk < 64; kk += 4) {                // m = gelu@W2 (K=64)
      v2f a = afrag(Sw, 68, kk, lane);
      g0 = wmma4(a, bfrag2(sW2[d], 33, kk, 0, lane), g0);
      g1 = wmma4(a, bfrag2(sW2[d], 33, kk, 16, lane), g1);
    }
    g0 = g0 + y0; g1 = g1 + y1;                         // residual
    if (d == 0) { f0 = g0; f1 = g1; } else { m0 = g0; m1 = g1; }
    __builtin_amdgcn_wave_barrier();
  }

  cstore(Sw, 68, 0, lane, f0);  cstore(Sw, 68, 16, lane, f1);   // [fwd | bwd]
  cstore(Sw, 68, 32, lane, m0); cstore(Sw, 68, 48, lane, m1);
  __builtin_amdgcn_wave_barrier();

  v8f o0, o1;
  { float b0 = sbp[col], b1 = sbp[col + 16];
#pragma unroll
    for (int i = 0; i < 8; ++i) { o0[i] = b0; o1[i] = b1; } }
#pragma unroll
  for (int kk = 0; kk < 64; kk += 4) {                  // out = [fwd,bwd]@Wpo
    v2f a = afrag(Sw, 68, kk, lane);
    o0 = wmma4(a, bfrag2(sWp, 33, kk, 0, lane), o0);
    o1 = wmma4(a, bfrag2(sWp, 33, kk, 16, lane), o1);
  }
  __builtin_amdgcn_wave_barrier();
  cstore(Sw, 36, 0, lane, o0);
  cstore(Sw, 36, 16, lane, o1);
  __builtin_amdgcn_wave_barrier();
#pragma unroll
  for (int i = 0; i < 4; ++i) {                         // coalesced NT output store
    int j = lane + 32 * i; int row = j >> 3, c4 = j & 7;
    v4f v = *((const v4f*)(Sw + row * 36) + c4);
    __builtin_nontemporal_store(v, (v4f*)(out + n0 * WD) + j);
  }
}

extern "C" void kernel_launch(void* const* d_in, const int* in_sizes, int n_in,
                              void* d_out, int out_size, void* d_ws, size_t ws_size,
                              hipStream_t stream) {
  (void)in_sizes; (void)n_in; (void)out_size; (void)ws_size;
  const float* e     = (const float*)d_in[0];
  const float* Wpi   = (const float*)d_in[1];
  const float* bpi   = (const float*)d_in[2];
  const float* Wpo   = (const float*)d_in[3];
  const float* bpo   = (const float*)d_in[4];
  const float* fWin  = (const float*)d_in[5];
  const float* fbin  = (const float*)d_in[6];
  const float* floga = (const float*)d_in[7];
  const float* fWout = (const float*)d_in[8];
  const float* fbout = (const float*)d_in[9];
  const float* fW1   = (const float*)d_in[10];
  const float* fb1   = (const float*)d_in[11];
  const float* fW2   = (const float*)d_in[12];
  const float* fb2   = (const float*)d_in[13];
  const float* bWin  = (const float*)d_in[14];
  const float* bbin  = (const float*)d_in[15];
  const float* bloga = (const float*)d_in[16];
  const float* bWout = (const float*)d_in[17];
  const float* bbout = (const float*)d_in[18];
  const float* bW1   = (const float*)d_in[19];
  const float* bb1   = (const float*)d_in[20];
  const float* bW2   = (const float*)d_in[21];
  const float* bb2   = (const float*)d_in[22];

  float* uf    = (float*)d_ws;                          // (B,T,H) fp32, becomes h_f
  float* ub    = uf + (size_t)WB * WT * WH;             // (B,T,H) fp32, becomes h_b
  float* carry = ub + (size_t)WB * WT * WH;             // chunk carries/prefixes
  float* out   = (float*)d_out;

  k1_proj_u<<<NTILES / WAVES, 256, 0, stream>>>(e, Wpi, bpi, fWin, fbin, bWin, bbin, uf, ub);
  k2a_localscan<<<(2 * WB * NCH * WH) / 256, 256, 0, stream>>>(floga, bloga, uf, ub, carry);
  k2b_chunkprefix<<<(2 * WB * WH) / 256, 256, 0, stream>>>(floga, bloga, carry);
  k2c_apply<<<2 * WB * NCH, 32, 0, stream>>>(floga, bloga, uf, ub, carry);
  k3_head<<<NTILES / WAVES, 256, 0, stream>>>(uf, ub, fWout, fbout, fW1, fb1, fW2, fb2,
                                              bWout, bbout, bW1, bb1, bW2, bb2, Wpo, bpo, out);
}